// Bert_GraphAttentionPrototype_44212393345172
// MI455X (gfx1250) — compile-verified
//
#include <hip/hip_runtime.h>
#include <hip/hip_bf16.h>
#include <math.h>

typedef __attribute__((ext_vector_type(2))) float v2f;
typedef __attribute__((ext_vector_type(8))) float v8f;

#if defined(__HIP_DEVICE_COMPILE__)
#if !__has_builtin(__builtin_amdgcn_wmma_f32_16x16x4_f32)
#error "missing __builtin_amdgcn_wmma_f32_16x16x4_f32 on this toolchain (device pass)"
#endif
#endif

#define KDIM 768     // inner dim for every GEMM in this problem
#define KB   16      // K-block staged in LDS (double buffered)
#define NKB  (KDIM / KB)
#define LSTR 20      // padded LDS row stride (floats): conflict-free, 80B rows (16B ok)
#define NS   32768
#define NP   1024
#define NA   256
#define FINF 0x7F800000u

// ---------------------------------------------------------------------------
// async copy of a 128 x KB tile (row-major src, ld = KDIM) into LDS.
// Each thread moves 2x16B via GLOBAL_LOAD_ASYNC_TO_LDS_B128 (ASYNCcnt).
// ---------------------------------------------------------------------------
__device__ __forceinline__ void async_tile(const float* __restrict__ src,
                                           float* dst, int kBase, int t) {
  const int lr = t >> 2;        // 0..63
  const int lc = (t & 3) << 2;  // 0,4,8,12
#pragma unroll
  for (int i = 0; i < 2; ++i) {
    const int row = lr + (i << 6);
    const float* g = src + (size_t)row * KDIM + kBase + lc;
    // low 32 bits of a generic shared pointer == LDS byte offset
    const unsigned l = (unsigned)(uintptr_t)(dst + row * LSTR + lc);
    asm volatile("global_load_async_to_lds_b128 %0, %1, off"
                 :: "v"(l), "v"(g) : "memory");
  }
}

__device__ __forceinline__ void wait_async_le4() {
  asm volatile("s_wait_asynccnt 0x4" ::: "memory");
}
__device__ __forceinline__ void wait_async_0() {
  asm volatile("s_wait_asynccnt 0x0" ::: "memory");
}

// ---------------------------------------------------------------------------
// 128x128 (M x N) fp32 GEMM block: acc += Ablk * Bblk^T, K = KDIM.
// Double-buffered async global->LDS pipeline; 8 waves; wave w owns
// rows [(w&3)*32,+32) x cols [(w>>2)*64,+64) as 2x4 accumulators of 16x16,
// driven by V_WMMA_F32_16X16X4_F32.
// ---------------------------------------------------------------------------
__device__ __forceinline__ void gemm_block_128(const float* __restrict__ Ag,
                                               const float* __restrict__ Bg,
                                               float* As0, float* Bs0,
                                               float* As1, float* Bs1,
                                               v8f acc[2][4]) {
  const int t = threadIdx.x;
  const int lane = t & 31;
  const int half = lane >> 4;
  const int l16 = lane & 15;
  const int w = t >> 5;
  const int wm = (w & 3) << 5;  // 0,32,64,96
  const int wn = (w >> 2) << 6; // 0,64

  // prologue: fill buffer 0 (4 async ops / wave outstanding)
  async_tile(Ag, As0, 0, t);
  async_tile(Bg, Bs0, 0, t);

  for (int kbi = 0; kbi < NKB; ++kbi) {
    float* Ac = (kbi & 1) ? As1 : As0;
    float* Bc = (kbi & 1) ? Bs1 : Bs0;
    float* An = (kbi & 1) ? As0 : As1;
    float* Bn = (kbi & 1) ? Bs0 : Bs1;
    if (kbi + 1 < NKB) {
      // issue next block's copies first, then retire current group:
      // async loads complete in order, so asynccnt<=4 frees buffer `cur`.
      async_tile(Ag, An, (kbi + 1) * KB, t);
      async_tile(Bg, Bn, (kbi + 1) * KB, t);
      wait_async_le4();
    } else {
      wait_async_0();
    }
    __syncthreads();   // current buffer visible to all waves
#pragma unroll
    for (int ks = 0; ks < KB; ks += 4) {
      v2f af[2], bf[4];
#pragma unroll
      for (int mi = 0; mi < 2; ++mi)
        af[mi] = *reinterpret_cast<const v2f*>(
            &Ac[(wm + mi * 16 + l16) * LSTR + ks + half * 2]);
#pragma unroll
      for (int ni = 0; ni < 4; ++ni)
        bf[ni] = *reinterpret_cast<const v2f*>(
            &Bc[(wn + ni * 16 + l16) * LSTR + ks + half * 2]);
#pragma unroll
      for (int mi = 0; mi < 2; ++mi)
#pragma unroll
        for (int ni = 0; ni < 4; ++ni)
          acc[mi][ni] = __builtin_amdgcn_wmma_f32_16x16x4_f32(
              false, af[mi], false, bf[ni], (short)0, acc[mi][ni], false, false);
    }
    __syncthreads();   // done reading `cur`; next iter may overwrite it
  }
}

#define DECLARE_TILES()              \
  __shared__ float As0[128 * LSTR]; \
  __shared__ float Bs0[128 * LSTR]; \
  __shared__ float As1[128 * LSTR]; \
  __shared__ float Bs1[128 * LSTR]

// ---------------------------------------------------------------------------
// workspace init: +inf min buffers, zero partials
// ---------------------------------------------------------------------------
__global__ void k_init(unsigned* __restrict__ minP, unsigned* __restrict__ minN,
                       float* __restrict__ pp) {
  const int i = blockIdx.x * blockDim.x + threadIdx.x;
  if (i < NS) minP[i] = FINF;
  if (i < NP) minN[i] = FINF;
  if (i < 64) pp[i] = 0.0f;
}

// ---------------------------------------------------------------------------
// squared row norms of z (32768x768) and prototypes (1024x768); 1 wave / row
// ---------------------------------------------------------------------------
__global__ void __launch_bounds__(256) k_norms(const float* __restrict__ z,
                                               const float* __restrict__ proto,
                                               float* __restrict__ zn2,
                                               float* __restrict__ pn2) {
  const int w = threadIdx.x >> 5, lane = threadIdx.x & 31;
  const int row = blockIdx.x * 8 + w;
  const float* src;
  float* dst;
  if (row < NS) { src = z + (size_t)row * KDIM;            dst = zn2 + row; }
  else          { src = proto + (size_t)(row - NS) * KDIM; dst = pn2 + (row - NS); }
  float s = 0.0f;
#pragma unroll
  for (int kk = 0; kk < 6; ++kk) {
    const float4 v = *reinterpret_cast<const float4*>(src + kk * 128 + lane * 4);
    s += v.x * v.x + v.y * v.y + v.z * v.z + v.w * v.w;
  }
#pragma unroll
  for (int m = 16; m; m >>= 1) s += __shfl_xor(s, m, 32);
  if (lane == 0) *dst = s;
}

// ---------------------------------------------------------------------------
// main tile kernel: G = z @ P^T block; fold into per-row / per-col d2 minima
// ---------------------------------------------------------------------------
__global__ void __launch_bounds__(256) k_zp(const float* __restrict__ z,
                                            const float* __restrict__ proto,
                                            const float* __restrict__ zn2,
                                            const float* __restrict__ pn2,
                                            unsigned* __restrict__ minP,
                                            unsigned* __restrict__ minN) {
  DECLARE_TILES();
  v8f acc[2][4] = {};
  const int rowBase = blockIdx.x * 128;
  const int colBase = blockIdx.y * 128;
  gemm_block_128(z + (size_t)rowBase * KDIM, proto + (size_t)colBase * KDIM,
                 As0, Bs0, As1, Bs1, acc);

  const int t = threadIdx.x, lane = t & 31, half = lane >> 4, l16 = lane & 15;
  const int w = t >> 5, wm = (w & 3) << 5, wn = (w >> 2) << 6;

  float pc[4];
#pragma unroll
  for (int ni = 0; ni < 4; ++ni) pc[ni] = pn2[colBase + wn + ni * 16 + l16];

  float colmin[4] = {INFINITY, INFINITY, INFINITY, INFINITY};
#pragma unroll
  for (int mi = 0; mi < 2; ++mi) {
#pragma unroll
    for (int r = 0; r < 8; ++r) {
      const int rowg = rowBase + wm + mi * 16 + half * 8 + r;
      const float zr = zn2[rowg];
      float rmin = INFINITY;
#pragma unroll
      for (int ni = 0; ni < 4; ++ni) {
        float d2 = zr + pc[ni] - 2.0f * acc[mi][ni][r];
        d2 = fmaxf(d2, 0.0f);
        rmin = fminf(rmin, d2);
        colmin[ni] = fminf(colmin[ni], d2);
      }
      // min over the 16 columns held by this half-wave
#pragma unroll
      for (int m = 8; m; m >>= 1) rmin = fminf(rmin, __shfl_xor(rmin, m, 32));
      if (l16 == 0) atomicMin(&minP[rowg], __float_as_uint(rmin));
    }
  }
#pragma unroll
  for (int ni = 0; ni < 4; ++ni) {
    const float c = fminf(colmin[ni], __shfl_xor(colmin[ni], 16, 32));
    if (half == 0)
      atomicMin(&minN[colBase + wn + ni * 16 + l16], __float_as_uint(c));
  }
}

// ---------------------------------------------------------------------------
// prototype-prototype: masked (j>i) sum of sqrt(d2) -> per-block partial
// ---------------------------------------------------------------------------
__global__ void __launch_bounds__(256) k_pp(const float* __restrict__ proto,
                                            const float* __restrict__ pn2,
                                            float* __restrict__ ppPartial) {
  DECLARE_TILES();
  v8f acc[2][4] = {};
  const int rowBase = blockIdx.x * 128;
  const int colBase = blockIdx.y * 128;
  gemm_block_128(proto + (size_t)rowBase * KDIM, proto + (size_t)colBase * KDIM,
                 As0, Bs0, As1, Bs1, acc);

  const int t = threadIdx.x, lane = t & 31, half = lane >> 4, l16 = lane & 15;
  const int w = t >> 5, wm = (w & 3) << 5, wn = (w >> 2) << 6;

  float pc[4];
#pragma unroll
  for (int ni = 0; ni < 4; ++ni) pc[ni] = pn2[colBase + wn + ni * 16 + l16];

  float s = 0.0f;
#pragma unroll
  for (int mi = 0; mi < 2; ++mi) {
#pragma unroll
    for (int r = 0; r < 8; ++r) {
      const int rowg = rowBase + wm + mi * 16 + half * 8 + r;
      const float pr = pn2[rowg];
#pragma unroll
      for (int ni = 0; ni < 4; ++ni) {
        const int colg = colBase + wn + ni * 16 + l16;
        if (colg > rowg) {
          const float d2 = pr + pc[ni] - 2.0f * acc[mi][ni][r];
          s += sqrtf(fmaxf(d2, 0.0f));
        }
      }
    }
  }
#pragma unroll
  for (int m = 16; m; m >>= 1) s += __shfl_xor(s, m, 32);
  if (lane == 0) As0[w] = s;   // reuse LDS (past final barrier of gemm)
  __syncthreads();
  if (t == 0) {
    float b = 0.0f;
#pragma unroll
    for (int i = 0; i < 8; ++i) b += As0[i];
    ppPartial[blockIdx.y * 8 + blockIdx.x] = b;  // gridDim.x == 8
  }
}

// ---------------------------------------------------------------------------
// encoded_{key,value}: out[p, a] = proto[p,:] . W[a,:] + bias[a]
// ---------------------------------------------------------------------------
__global__ void __launch_bounds__(256) k_enc(const float* __restrict__ proto,
                                             const float* __restrict__ W,
                                             const float* __restrict__ bias,
                                             float* __restrict__ out) {
  DECLARE_TILES();
  v8f acc[2][4] = {};
  const int rowBase = blockIdx.x * 128;
  const int colBase = blockIdx.y * 128;
  gemm_block_128(proto + (size_t)rowBase * KDIM, W + (size_t)colBase * KDIM,
                 As0, Bs0, As1, Bs1, acc);

  const int t = threadIdx.x, lane = t & 31, half = lane >> 4, l16 = lane & 15;
  const int w = t >> 5, wm = (w & 3) << 5, wn = (w >> 2) << 6;

  float bc[4];
#pragma unroll
  for (int ni = 0; ni < 4; ++ni) bc[ni] = bias[colBase + wn + ni * 16 + l16];

#pragma unroll
  for (int mi = 0; mi < 2; ++mi)
#pragma unroll
    for (int r = 0; r < 8; ++r) {
      const int rowg = rowBase + wm + mi * 16 + half * 8 + r;
#pragma unroll
      for (int ni = 0; ni < 4; ++ni) {
        const int colg = colBase + wn + ni * 16 + l16;
        out[(size_t)rowg * NA + colg] = acc[mi][ni][r] + bc[ni];
      }
    }
}

// ---------------------------------------------------------------------------
// finalize the three scalar losses
// ---------------------------------------------------------------------------
__global__ void __launch_bounds__(1024) k_final(const unsigned* __restrict__ minP,
                                                const unsigned* __restrict__ minN,
                                                const float* __restrict__ pp,
                                                float* __restrict__ outLoss) {
  __shared__ float red[1024];
  const int t = threadIdx.x;

  auto blockSum = [&](float v) -> float {
    red[t] = v;
    __syncthreads();
    for (int off = 512; off; off >>= 1) {
      if (t < off) red[t] += red[t + off];
      __syncthreads();
    }
    const float r = red[0];
    __syncthreads();
    return r;
  };

  float s1 = sqrtf(__uint_as_float(minN[t]));  // t in [0,1024)
  float s2 = 0.0f;
  for (int i = t; i < NS; i += 1024) s2 += sqrtf(__uint_as_float(minP[i]));
  float s3 = (t < 64) ? pp[t] : 0.0f;

  const float S1 = blockSum(s1);
  const float S2 = blockSum(s2);
  const float S3 = blockSum(s3);
  if (t == 0) {
    outLoss[0] = S1 / (float)NP;                       // loss_zp
    outLoss[1] = S2 / (float)NS;                       // loss_pz
    outLoss[2] = -S3 / (float)(NP * (NP + 1) / 2);     // loss_pp (denom 524800)
  }
}

// ---------------------------------------------------------------------------
// host-side launch: in order z, proto, Wk, bk, Wv, bv ; out = key|value|losses
// ---------------------------------------------------------------------------
extern "C" void kernel_launch(void* const* d_in, const int* in_sizes, int n_in,
                              void* d_out, int out_size, void* d_ws, size_t ws_size,
                              hipStream_t stream) {
  const float* z     = (const float*)d_in[0];
  const float* proto = (const float*)d_in[1];
  const float* Wk    = (const float*)d_in[2];
  const float* bk    = (const float*)d_in[3];
  const float* Wv    = (const float*)d_in[4];
  const float* bv    = (const float*)d_in[5];
  float* out = (float*)d_out;
  float* ws  = (float*)d_ws;

  // workspace layout (floats): zn2[32768] pn2[1024] minP[32768] minN[1024] pp[64]
  float*    zn2  = ws;
  float*    pn2  = ws + NS;
  unsigned* minP = (unsigned*)(ws + NS + NP);
  unsigned* minN = (unsigned*)(ws + 2 * NS + NP);
  float*    pp   = ws + 2 * NS + 2 * NP;

  k_init <<<dim3(NS / 256), dim3(256), 0, stream>>>(minP, minN, pp);
  k_norms<<<dim3((NS + NP) / 8), dim3(256), 0, stream>>>(z, proto, zn2, pn2);
  k_zp   <<<dim3(NS / 128, NP / 128), dim3(256), 0, stream>>>(z, proto, zn2, pn2,
                                                              minP, minN);
  k_pp   <<<dim3(NP / 128, NP / 128), dim3(256), 0, stream>>>(proto, pn2, pp);
  k_enc  <<<dim3(NP / 128, NA / 128), dim3(256), 0, stream>>>(proto, Wk, bk, out);
  k_enc  <<<dim3(NP / 128, NA / 128), dim3(256), 0, stream>>>(proto, Wv, bv,
                                                              out + NP * NA);
  k_final<<<dim3(1), dim3(1024), 0, stream>>>(minP, minN, pp, out + 2 * NP * NA);
}